// GCNFeatureExtractor_29884382445799
// MI455X (gfx1250) — compile-verified
//
#include <hip/hip_runtime.h>
#include <math.h>

#define D 64
#define G 256

typedef float v2f __attribute__((ext_vector_type(2)));
typedef float v8f __attribute__((ext_vector_type(8)));

// ---------------- degree / norm ----------------
__global__ void k_init_deg(float* deg, int n) {
    int i = blockIdx.x * blockDim.x + threadIdx.x;
    if (i < n) deg[i] = 1.0f;  // self loop
}
__global__ void k_deg_accum(const int* __restrict__ dst, float* deg, int e) {
    int i = blockIdx.x * blockDim.x + threadIdx.x;
    if (i < e) atomicAdd(&deg[dst[i]], 1.0f);
}
__global__ void k_deg_rsqrt(float* deg, int n) {
    int i = blockIdx.x * blockDim.x + threadIdx.x;
    if (i < n) deg[i] = rsqrtf(deg[i]);
}

// ---------------- GEMM: T = Hin @ W + bias  (WMMA f32 16x16x4) ----------------
// Each wave: 16 rows x 64 cols (4 accumulator tiles). Block = 8 waves = 128 rows.
// W is staged in LDS pre-paired along K: Wp[k/2][n] = {W[k][n], W[k+1][n]} so a
// lane's B-fragment is a single aligned ds_load_b64 (no register repacking).
__global__ void __launch_bounds__(256) k_gemm_bias(const float* __restrict__ Hin,
                                                   const float* __restrict__ W,
                                                   const float* __restrict__ bias,
                                                   float* __restrict__ T, int nrows) {
    __shared__ float2 Wp[(D / 2) * D];  // 32 x 64 float2 = 16 KB
    for (int i = threadIdx.x; i < (D / 2) * D; i += 256) {
        int p = i >> 6, n = i & 63;
        Wp[i] = make_float2(W[(2 * p) * D + n], W[(2 * p + 1) * D + n]);
    }
    __syncthreads();

    const int lane  = threadIdx.x & 31;
    const int lrow  = lane & 15;   // row within A tile / col within B,C tiles
    const int khalf = lane >> 4;   // 0 or 1: selects K pair {0,1} vs {2,3}
    const int wave  = threadIdx.x >> 5;

    long long m0 = ((long long)blockIdx.x * 8 + wave) * 16;
    long long arow = m0 + lrow;
    if (arow > (long long)nrows - 1) arow = (long long)nrows - 1;  // clamp: keep EXEC all-ones
    const float* A = Hin + arow * D;

    v8f acc[4] = {};
    for (int k = 0; k < D; k += 4) {
        int kb = k + khalf * 2;
        v2f a;
        a.x = A[kb];
        a.y = A[kb + 1];
        int prow = (k >> 1) + khalf;  // = kb/2 (kb always even)
#pragma unroll
        for (int t = 0; t < 4; ++t) {
            int n = t * 16 + lrow;
            float2 bw = Wp[prow * D + n];
            v2f b;
            b.x = bw.x;
            b.y = bw.y;
            acc[t] = __builtin_amdgcn_wmma_f32_16x16x4_f32(
                false, a, false, b, (short)0, acc[t], false, false);
        }
    }
#pragma unroll
    for (int t = 0; t < 4; ++t) {
        int n = t * 16 + lrow;
        float bn = bias[n];
#pragma unroll
        for (int v = 0; v < 8; ++v) {
            long long m = m0 + v + 8 * khalf;  // C/D layout: VGPR v -> M = v + 8*(lane>=16)
            if (m < nrows) T[m * D + n] = acc[t][v] + bn;
        }
    }
}

// ---------------- aggregation ----------------
// Self-loop term initializes the output buffer: O[i] = T[i] * dinv[i]^2
__global__ void k_selfloop(const float* __restrict__ T, const float* __restrict__ dinv,
                           float* __restrict__ O, int nquad) {  // nquad = N*16
    int i = blockIdx.x * blockDim.x + threadIdx.x;
    if (i >= nquad) return;
    int node = i >> 4, q = i & 15;
    float dv = dinv[node];
    float nrm = dv * dv;
    float4 v = ((const float4*)T)[(long long)node * 16 + q];
    float4 o = {v.x * nrm, v.y * nrm, v.z * nrm, v.w * nrm};
    ((float4*)O)[(long long)node * 16 + q] = o;
}

// Per-edge scatter: one thread per (edge, 4 features)
__global__ void k_scatter(const float* __restrict__ T, const int* __restrict__ src,
                          const int* __restrict__ dst, const float* __restrict__ dinv,
                          float* O, long long nwork) {  // nwork = E*16
    long long i = (long long)blockIdx.x * blockDim.x + threadIdx.x;
    if (i >= nwork) return;
    int e = (int)(i >> 4), q = (int)(i & 15);
    int s = src[e], d = dst[e];
    float nrm = dinv[s] * dinv[d];
    float4 v = ((const float4*)T)[(long long)s * 16 + q];
    float* o = O + (long long)d * D + q * 4;
    atomicAdd(o + 0, v.x * nrm);
    atomicAdd(o + 1, v.y * nrm);
    atomicAdd(o + 2, v.z * nrm);
    atomicAdd(o + 3, v.w * nrm);
}

// ---------------- batchnorm ----------------
__global__ void k_fill(float* p, float val, int n) {
    int i = blockIdx.x * blockDim.x + threadIdx.x;
    if (i < n) p[i] = val;
}

// stats[0:64]=sum, stats[64:128]=sumsq
__global__ void k_stats(const float* __restrict__ H, float* stats, int nrows) {
    __shared__ float ssum[256], ssq[256];
    int f = threadIdx.x & 63;
    int rowOff = threadIdx.x >> 6;  // 0..3
    float s = 0.f, s2 = 0.f;
    for (long long r = (long long)blockIdx.x * 4 + rowOff; r < nrows;
         r += (long long)gridDim.x * 4) {
        float v = H[r * D + f];
        s += v;
        s2 += v * v;
    }
    ssum[threadIdx.x] = s;
    ssq[threadIdx.x] = s2;
    __syncthreads();
    if (threadIdx.x < 64) {
        s  = ssum[f] + ssum[f + 64] + ssum[f + 128] + ssum[f + 192];
        s2 = ssq[f]  + ssq[f + 64]  + ssq[f + 128]  + ssq[f + 192];
        atomicAdd(&stats[f], s);
        atomicAdd(&stats[64 + f], s2);
    }
}

// stats[128:192]=scale, stats[192:256]=shift
__global__ void k_bn_finalize(const float* __restrict__ g, const float* __restrict__ be,
                              float* stats, float invN) {
    int f = threadIdx.x;  // 64 threads
    float mean = stats[f] * invN;
    float var  = stats[64 + f] * invN - mean * mean;
    float sc   = g[f] * rsqrtf(var + 1e-5f);
    stats[128 + f] = sc;
    stats[192 + f] = be[f] - mean * sc;
}

__global__ void k_bn_apply(float* H, const float* __restrict__ stats, int relu, long long n) {
    long long i = (long long)blockIdx.x * blockDim.x + threadIdx.x;
    if (i >= n) return;
    int f = (int)(i & 63);
    float v = H[i] * stats[128 + f] + stats[192 + f];
    if (relu) v = fmaxf(v, 0.f);
    H[i] = v;
}

// ---------------- pooling ----------------
__device__ inline void atomicMaxF(float* addr, float val) {
    if (val >= 0.f)
        atomicMax((int*)addr, __float_as_int(val));
    else
        atomicMin((unsigned int*)addr, (unsigned int)__float_as_int(val));
}

__global__ void k_pool_accum(const float* __restrict__ H, const int* __restrict__ batch,
                             float* psum, float* pmax, float* pcnt, long long n) {
    long long i = (long long)blockIdx.x * blockDim.x + threadIdx.x;
    if (i >= n) return;
    int node = (int)(i >> 6), f = (int)(i & 63);
    int b = batch[node];
    float v = H[i];
    atomicAdd(&psum[b * D + f], v);
    atomicMaxF(&pmax[b * D + f], v);
    if (f == 0) atomicAdd(&pcnt[b], 1.0f);
}

__global__ void k_pool_final(const float* __restrict__ psum, const float* __restrict__ pmax,
                             const float* __restrict__ pcnt, float* out) {
    int i = blockIdx.x * blockDim.x + threadIdx.x;
    if (i >= G * D) return;
    int g = i >> 6;
    float c = pcnt[g];
    float mean = psum[i] / fmaxf(c, 1.0f);
    float mx = (c > 0.f) ? pmax[i] : 0.f;
    out[i] = mean + mx;
}

// ---------------- driver ----------------
extern "C" void kernel_launch(void* const* d_in, const int* in_sizes, int n_in,
                              void* d_out, int out_size, void* d_ws, size_t ws_size,
                              hipStream_t stream) {
    const float* x     = (const float*)d_in[0];
    const int*   ei    = (const int*)d_in[1];
    const int*   batch = (const int*)d_in[2];
    const float* Wl[3]  = {(const float*)d_in[3], (const float*)d_in[7],  (const float*)d_in[11]};
    const float* bl[3]  = {(const float*)d_in[4], (const float*)d_in[8],  (const float*)d_in[12]};
    const float* gl[3]  = {(const float*)d_in[5], (const float*)d_in[9],  (const float*)d_in[13]};
    const float* bel[3] = {(const float*)d_in[6], (const float*)d_in[10], (const float*)d_in[14]};

    const int N = in_sizes[0] / D;
    const int E = in_sizes[1] / 2;
    const int* src = ei;
    const int* dst = ei + E;

    char* ws = (char*)d_ws;
    size_t off = 0;
    auto alloc = [&](size_t bytes) -> void* {
        off = (off + 255) & ~(size_t)255;
        void* p = ws + off;
        off += bytes;
        return p;
    };
    float* dinv  = (float*)alloc((size_t)N * 4);
    float* T     = (float*)alloc((size_t)N * D * 4);  // GEMM output
    float* Hb    = (float*)alloc((size_t)N * D * 4);  // aggregated / normalized features
    float* stats = (float*)alloc(256 * 4);
    float* psum  = (float*)alloc((size_t)G * D * 4);
    float* pmax  = (float*)alloc((size_t)G * D * 4);
    float* pcnt  = (float*)alloc((size_t)G * 4);

    // degree with self loops -> dinv
    k_init_deg<<<(N + 255) / 256, 256, 0, stream>>>(dinv, N);
    k_deg_accum<<<(E + 255) / 256, 256, 0, stream>>>(dst, dinv, E);
    k_deg_rsqrt<<<(N + 255) / 256, 256, 0, stream>>>(dinv, N);

    const long long nScatter = (long long)E * 16;
    const int       nSelf    = N * 16;
    const long long nElem    = (long long)N * D;
    const int gemmBlocks     = (N + 127) / 128;
    const float invN         = 1.0f / (float)N;

    const float* hin = x;
    for (int l = 0; l < 3; ++l) {
        k_gemm_bias<<<gemmBlocks, 256, 0, stream>>>(hin, Wl[l], bl[l], T, N);
        k_selfloop<<<(nSelf + 255) / 256, 256, 0, stream>>>(T, dinv, Hb, nSelf);
        k_scatter<<<(int)((nScatter + 255) / 256), 256, 0, stream>>>(T, src, dst, dinv, Hb, nScatter);
        k_fill<<<1, 128, 0, stream>>>(stats, 0.f, 128);
        k_stats<<<1024, 256, 0, stream>>>(Hb, stats, N);
        k_bn_finalize<<<1, 64, 0, stream>>>(gl[l], bel[l], stats, invN);
        k_bn_apply<<<(int)((nElem + 255) / 256), 256, 0, stream>>>(Hb, stats, (l < 2) ? 1 : 0, nElem);
        hin = Hb;
    }

    // pooling
    k_fill<<<(G * D + 255) / 256, 256, 0, stream>>>(psum, 0.f, G * D);
    k_fill<<<(G * D + 255) / 256, 256, 0, stream>>>(pmax, -INFINITY, G * D);
    k_fill<<<1, 256, 0, stream>>>(pcnt, 0.f, G);
    k_pool_accum<<<(int)((nElem + 255) / 256), 256, 0, stream>>>(Hb, batch, psum, pmax, pcnt, nElem);
    k_pool_final<<<(G * D + 255) / 256, 256, 0, stream>>>(psum, pmax, pcnt, (float*)d_out);
}